// Nequip_13331578487442
// MI455X (gfx1250) — compile-verified
//
#include <hip/hip_runtime.h>
#include <hip/hip_bf16.h>
#include <math.h>

typedef float v2f __attribute__((ext_vector_type(2)));
typedef float v8f __attribute__((ext_vector_type(8)));

#define N_NODES 20000
#define N_EDGES 400000
#define C_CH    16
#define SH_DIM  9
#define NB      8
#define HDIM    64
#define CUTOFF_R 5.0f
#define DEG_NORM 0.22360679774997896f   /* 1/sqrt(E/N) = 1/sqrt(20) */
#define LN2F     0.6931471805599453f

__device__ __forceinline__ float ssp(float v) {
    // softplus(v) - ln2, numerically stable
    return fmaxf(v, 0.f) + log1pf(expf(-fabsf(v))) - LN2F;
}
__device__ __forceinline__ float sigmoidf(float v) {
    return 1.f / (1.f + expf(-v));
}

// ---------------------------------------------------------------------------
// Edge precompute: Bessel-with-envelope basis (E x 8) and spherical harmonics
// Y (E x 9). env is already 0 for r >= cutoff (identical condition to mask).
// ---------------------------------------------------------------------------
__global__ void edge_pre_kernel(const float* __restrict__ edge_diff,
                                const float* __restrict__ edge_dist,
                                float* __restrict__ ebasis,
                                float* __restrict__ Yv) {
    int e = blockIdx.x * 256 + threadIdx.x;
    if (e >= N_EDGES) return;

    float r  = edge_dist[e];
    float u  = r / CUTOFF_R;
    float u2 = u * u, u3 = u2 * u, u6 = u3 * u3, u7 = u6 * u, u8 = u7 * u;
    // POWER=6: (p+1)(p+2)/2 = 28, p(p+2) = 48, p(p+1)/2 = 21
    float env  = (u < 1.f) ? (1.f - 28.f * u6 + 48.f * u7 - 21.f * u8) : 0.f;
    float pref = sqrtf(2.f / CUTOFF_R) / r * env;
    float base = 3.14159265358979323846f * r / CUTOFF_R;
    float* eb = ebasis + e * NB;
#pragma unroll
    for (int n = 1; n <= NB; ++n)
        eb[n - 1] = pref * sinf((float)n * base);

    float dx = edge_diff[e * 3 + 0];
    float dy = edge_diff[e * 3 + 1];
    float dz = edge_diff[e * 3 + 2];
    float inv = rsqrtf(dx * dx + dy * dy + dz * dz);
    float X = dx * inv, Y = dy * inv, Z = dz * inv;
    const float s3 = 1.7320508075688772f;
    const float s5 = 2.2360679774997896f;
    const float s15 = 3.872983346207417f;
    float* yo = Yv + e * SH_DIM;
    yo[0] = 1.f;
    yo[1] = s3 * X;
    yo[2] = s3 * Y;
    yo[3] = s3 * Z;
    yo[4] = s15 * X * Y;
    yo[5] = s15 * Y * Z;
    yo[6] = 0.5f * s5 * (2.f * Z * Z - X * X - Y * Y);
    yo[7] = s15 * X * Z;
    yo[8] = 0.5f * s15 * (X * X - Y * Y);
}

// ---------------------------------------------------------------------------
// x init: x[n][c][0] = W_embed[species[n]][c], other SH components zero.
// ---------------------------------------------------------------------------
__global__ void node_init_kernel(const int* __restrict__ species,
                                 const float* __restrict__ W_embed,
                                 float* __restrict__ x) {
    int t = blockIdx.x * 256 + threadIdx.x;   // t = n*16 + c
    if (t >= N_NODES * C_CH) return;
    int n = t >> 4, c = t & 15;
    float* xr = x + t * SH_DIM;
    xr[0] = W_embed[species[n] * C_CH + c];
#pragma unroll
    for (int m = 1; m < SH_DIM; ++m) xr[m] = 0.f;
}

// ---------------------------------------------------------------------------
// Fused per-edge layer: WMMA f32 16x16x4 radial MLP + message + atomic scatter.
// Each wave owns 16 edges. 8 waves / block => 128 edges / block.
//   GEMM1: (16x8)  @ W1 (8x64)  -> H1 (16x64), ssp, staged in LDS
//   GEMM2: (16x64) @ W2 (64x32) -> R  (16x32)  (R1 | R2), *deg_norm -> LDS
//   Scatter: msg[e,c,m] = mask*(R1*x[src,c,0]*Y[e,m] + R2*x[src,c,m])
//            atomicAdd into agg[dst,c,m]
// f32 WMMA operand layouts (ISA 7.12.2):
//   A 16x4:  lane l -> M = l%16, VGPR0/1 = K = 2*(l/16) (+1)
//   B 4x16:  lane l -> N = l%16, VGPR0/1 = K = 2*(l/16) (+1)
//   C/D:     VGPR v, lane l -> row v + 8*(l/16), col l%16
// ---------------------------------------------------------------------------
__global__ void edge_layer_kernel(const float* __restrict__ ebasis,
                                  const float* __restrict__ Yv,
                                  const int*   __restrict__ edge_idx,
                                  const float* __restrict__ edge_dist,
                                  const float* __restrict__ x,
                                  const float* __restrict__ W1,
                                  const float* __restrict__ W2,
                                  float* __restrict__ agg) {
    __shared__ float h1s[8][16][HDIM];  // per-wave ssp(H1) tiles (32 KB)
    __shared__ float rs[8][16][32];     // per-wave R tiles       (16 KB)

    const int lane  = threadIdx.x & 31;
    const int wave  = threadIdx.x >> 5;
    const int M     = lane & 15;        // row within 16-edge tile / col N
    const int half  = lane >> 4;        // 0 or 1
    const int ebase = (blockIdx.x * 8 + wave) * 16;

    // ---- GEMM1: ebasis(16x8) @ W1(8x64) ----
    const float* ebrow = ebasis + (ebase + M) * NB;
    v2f aK0 = { ebrow[2 * half],     ebrow[2 * half + 1] };      // K = 0..3
    v2f aK1 = { ebrow[4 + 2 * half], ebrow[5 + 2 * half] };      // K = 4..7

#pragma unroll
    for (int n = 0; n < 4; ++n) {
        const int col = n * 16 + M;
        v8f c = {0.f, 0.f, 0.f, 0.f, 0.f, 0.f, 0.f, 0.f};
        v2f b0 = { W1[(2 * half) * HDIM + col], W1[(2 * half + 1) * HDIM + col] };
        c = __builtin_amdgcn_wmma_f32_16x16x4_f32(false, aK0, false, b0,
                                                  (short)0, c, false, false);
        v2f b1 = { W1[(4 + 2 * half) * HDIM + col], W1[(5 + 2 * half) * HDIM + col] };
        c = __builtin_amdgcn_wmma_f32_16x16x4_f32(false, aK1, false, b1,
                                                  (short)0, c, false, false);
#pragma unroll
        for (int v = 0; v < 8; ++v)
            h1s[wave][v + 8 * half][col] = ssp(c[v]);
    }
    __syncthreads();

    // ---- GEMM2: ssp(H1)(16x64) @ W2(64x32) ----
    v8f cr0 = {0.f, 0.f, 0.f, 0.f, 0.f, 0.f, 0.f, 0.f};
    v8f cr1 = {0.f, 0.f, 0.f, 0.f, 0.f, 0.f, 0.f, 0.f};
#pragma unroll
    for (int kk = 0; kk < 16; ++kk) {
        const int k0 = 4 * kk + 2 * half;
        v2f a = { h1s[wave][M][k0], h1s[wave][M][k0 + 1] };
        v2f b0 = { W2[k0 * 32 + M],      W2[(k0 + 1) * 32 + M] };
        cr0 = __builtin_amdgcn_wmma_f32_16x16x4_f32(false, a, false, b0,
                                                    (short)0, cr0, false, false);
        v2f b1 = { W2[k0 * 32 + 16 + M], W2[(k0 + 1) * 32 + 16 + M] };
        cr1 = __builtin_amdgcn_wmma_f32_16x16x4_f32(false, a, false, b1,
                                                    (short)0, cr1, false, false);
    }
#pragma unroll
    for (int v = 0; v < 8; ++v) {
        const int row = v + 8 * half;
        rs[wave][row][M]      = cr0[v] * DEG_NORM;   // R1
        rs[wave][row][16 + M] = cr1[v] * DEG_NORM;   // R2
    }
    __syncthreads();

    // ---- message + scatter: lane handles edge M, channels 8*half..8*half+7 ----
    const int e   = M;
    const int eg  = ebase + e;
    const int src = edge_idx[2 * eg + 0];
    const int dst = edge_idx[2 * eg + 1];
    const float mk = (edge_dist[eg] < CUTOFF_R) ? 1.f : 0.f;
    float Ye[SH_DIM];
#pragma unroll
    for (int m = 0; m < SH_DIM; ++m) Ye[m] = Yv[eg * SH_DIM + m];

    const float* xs = x   + (src * C_CH + 8 * half) * SH_DIM;
    float*       ad = agg + (dst * C_CH + 8 * half) * SH_DIM;
#pragma unroll
    for (int j = 0; j < 8; ++j) {
        const float R1 = rs[wave][e][8 * half + j] * mk;
        const float R2 = rs[wave][e][16 + 8 * half + j] * mk;
        const float* xr = xs + j * SH_DIM;
        float*       ar = ad + j * SH_DIM;
        const float xj0 = xr[0];
#pragma unroll
        for (int m = 0; m < SH_DIM; ++m)
            atomicAdd(ar + m, R1 * xj0 * Ye[m] + R2 * xr[m]);
    }
}

// ---------------------------------------------------------------------------
// Per-node update: channel mix per SH block (Wmix), gate (Wg), new x.
// 256 threads = 16 nodes x 16 output channels d.
// ---------------------------------------------------------------------------
__global__ void node_layer_kernel(const float* __restrict__ agg,
                                  const float* __restrict__ Wmix,
                                  const float* __restrict__ Wg,
                                  float* __restrict__ x) {
    __shared__ float sS[16][16];      // s = out[:, :, 0] per local node
    const int t  = threadIdx.x;
    const int d  = t & 15;
    const int nl = t >> 4;
    const int n  = blockIdx.x * 16 + nl;

    const float* ag = agg + n * C_CH * SH_DIM;
    float out[SH_DIM];
#pragma unroll
    for (int m = 0; m < SH_DIM; ++m) out[m] = 0.f;

#pragma unroll
    for (int c = 0; c < C_CH; ++c) {
        const float w0 = Wmix[0 * 256 + c * 16 + d];   // l = 0 block
        const float w1 = Wmix[1 * 256 + c * 16 + d];   // l = 1 block
        const float w2 = Wmix[2 * 256 + c * 16 + d];   // l = 2 block
        const float* a = ag + c * SH_DIM;
        out[0] += a[0] * w0;
#pragma unroll
        for (int m = 1; m < 4; ++m) out[m] += a[m] * w1;
#pragma unroll
        for (int m = 4; m < 9; ++m) out[m] += a[m] * w2;
    }

    const float s = out[0];
    sS[nl][d] = s;
    __syncthreads();

    float g1a = 0.f, g2a = 0.f;
#pragma unroll
    for (int c = 0; c < C_CH; ++c) {
        const float sc = sS[nl][c];
        g1a += sc * Wg[c * 32 + d];
        g2a += sc * Wg[c * 32 + 16 + d];
    }
    const float g1 = sigmoidf(g1a);
    const float g2 = sigmoidf(g2a);

    float* xr = x + (n * C_CH + d) * SH_DIM;
    xr[0] = ssp(s);
#pragma unroll
    for (int m = 1; m < 4; ++m) xr[m] = out[m] * g1;
#pragma unroll
    for (int m = 4; m < 9; ++m) xr[m] = out[m] * g2;
}

// ---------------------------------------------------------------------------
// Readout: out[n] = ssp(x[n,:,0] @ Wr1) @ wr2
// ---------------------------------------------------------------------------
__global__ void readout_kernel(const float* __restrict__ x,
                               const float* __restrict__ Wr1,
                               const float* __restrict__ wr2,
                               float* __restrict__ out) {
    int n = blockIdx.x * 256 + threadIdx.x;
    if (n >= N_NODES) return;
    float x0[C_CH];
#pragma unroll
    for (int c = 0; c < C_CH; ++c) x0[c] = x[(n * C_CH + c) * SH_DIM];
    float acc = 0.f;
    for (int h = 0; h < HDIM; ++h) {
        float hv = 0.f;
#pragma unroll
        for (int c = 0; c < C_CH; ++c) hv += x0[c] * Wr1[c * HDIM + h];
        acc += ssp(hv) * wr2[h];
    }
    out[n] = acc;
}

// ---------------------------------------------------------------------------
extern "C" void kernel_launch(void* const* d_in, const int* in_sizes, int n_in,
                              void* d_out, int out_size, void* d_ws, size_t ws_size,
                              hipStream_t stream) {
    const int*   species   = (const int*)  d_in[0];
    const int*   edge_idx  = (const int*)  d_in[1];
    const float* edge_diff = (const float*)d_in[2];
    const float* edge_dist = (const float*)d_in[3];
    const float* W_embed   = (const float*)d_in[4];
    const float* W1        = (const float*)d_in[5];  // (3, 8, 64)
    const float* W2        = (const float*)d_in[6];  // (3, 64, 32)
    const float* Wmix      = (const float*)d_in[7];  // (3, 3, 16, 16)
    const float* Wg        = (const float*)d_in[8];  // (3, 16, 32)
    const float* Wr1       = (const float*)d_in[9];  // (16, 64)
    const float* wr2       = (const float*)d_in[10]; // (64,)
    float* out = (float*)d_out;

    float* ws     = (float*)d_ws;
    float* ebasis = ws;                              // E*8
    float* Yv     = ebasis + (size_t)N_EDGES * NB;   // E*9
    float* x      = Yv + (size_t)N_EDGES * SH_DIM;   // N*144
    float* agg    = x + (size_t)N_NODES * C_CH * SH_DIM; // N*144

    edge_pre_kernel<<<(N_EDGES + 255) / 256, 256, 0, stream>>>(edge_diff, edge_dist,
                                                               ebasis, Yv);
    node_init_kernel<<<(N_NODES * C_CH + 255) / 256, 256, 0, stream>>>(species, W_embed, x);

    for (int i = 0; i < 3; ++i) {
        hipMemsetAsync(agg, 0, (size_t)N_NODES * C_CH * SH_DIM * sizeof(float), stream);
        edge_layer_kernel<<<N_EDGES / 128, 256, 0, stream>>>(
            ebasis, Yv, edge_idx, edge_dist, x,
            W1 + (size_t)i * NB * HDIM,
            W2 + (size_t)i * HDIM * 32,
            agg);
        node_layer_kernel<<<N_NODES / 16, 256, 0, stream>>>(
            agg,
            Wmix + (size_t)i * 3 * C_CH * C_CH,
            Wg   + (size_t)i * C_CH * 32,
            x);
    }
    readout_kernel<<<(N_NODES + 255) / 256, 256, 0, stream>>>(x, Wr1, wr2, out);
}